// GATModel_55490977464422
// MI455X (gfx1250) — compile-verified
//
#include <hip/hip_runtime.h>
#include <math.h>

#define N_NODES 50000
#define N_EDGES 600000
#define IN_DIM  128
#define HEADS   4
#define OUT_DIM 32
#define HD      (HEADS * OUT_DIM)           // 128
#define E_AUG   (2 * N_EDGES + N_NODES)     // 1,250,000

typedef float v2f __attribute__((ext_vector_type(2)));
typedef float v8f __attribute__((ext_vector_type(8)));

// ---------------------------------------------------------------------------
// Kernel 1: proj = x @ W1 via V_WMMA_F32_16X16X4_F32 (fp32 matrix cores).
// One wave per 16x16 output tile. 50000 rows = 3125 tiles * 16, 128 cols = 8 tiles.
// ---------------------------------------------------------------------------
__global__ __launch_bounds__(32)
void gemm_proj_kernel(const float* __restrict__ x,
                      const float* __restrict__ W1,
                      float* __restrict__ proj) {
  const int tile  = blockIdx.x;
  const int tileM = tile >> 3;     // 0..3124
  const int tileN = tile & 7;      // 0..7
  const int lane  = threadIdx.x;   // 0..31
  const int l     = lane & 15;
  const int half  = lane >> 4;

  const int rowA = tileM * 16 + l;        // A-frag: M = lane%16
  const int colB = tileN * 16 + l;        // B-frag: N = lane%16

  v8f c = {};
  for (int k = 0; k < IN_DIM; k += 4) {
    const int ka = k + half * 2;          // lanes 0-15: K=k,k+1 ; lanes 16-31: K=k+2,k+3
    v2f a, b;
    a.x = x[rowA * IN_DIM + ka];
    a.y = x[rowA * IN_DIM + ka + 1];
    b.x = W1[ka * HD + colB];
    b.y = W1[(ka + 1) * HD + colB];
    // 8 args: (neg_a, A, neg_b, B, c_mod, C, reuse_a, reuse_b)
    c = __builtin_amdgcn_wmma_f32_16x16x4_f32(false, a, false, b, (short)0, c,
                                              false, false);
  }
  // C/D layout: VGPR r -> M = r + 8*half, N = lane%16
  #pragma unroll
  for (int r = 0; r < 8; ++r) {
    const int row = tileM * 16 + r + half * 8;
    proj[row * HD + colB] = c[r];
  }
}

// ---------------------------------------------------------------------------
// Kernel 2: per-node attention logits alpha_src / alpha_dst  [N, H]
// ---------------------------------------------------------------------------
__global__ void alpha_kernel(const float* __restrict__ proj,
                             const float* __restrict__ a_src,
                             const float* __restrict__ a_dst,
                             float* __restrict__ alpha_src,
                             float* __restrict__ alpha_dst) {
  const int idx = blockIdx.x * blockDim.x + threadIdx.x;   // n*HEADS + h
  if (idx >= N_NODES * HEADS) return;
  const int n = idx >> 2;
  const int h = idx & 3;
  const float* p  = proj  + (size_t)n * HD + h * OUT_DIM;
  const float* as = a_src + h * OUT_DIM;
  const float* ad = a_dst + h * OUT_DIM;
  float ss = 0.f, sd = 0.f;
  #pragma unroll
  for (int d = 0; d < OUT_DIM; ++d) {
    const float v = p[d];
    ss += v * as[d];
    sd += v * ad[d];
  }
  alpha_src[idx] = ss;
  alpha_dst[idx] = sd;
}

// ---------------------------------------------------------------------------
// Kernel 3: init accumulators (accum=0, segmax=-inf-ish, denom=0)
// ---------------------------------------------------------------------------
__global__ void init_kernel(float* __restrict__ accum,
                            float* __restrict__ segmax,
                            float* __restrict__ denom) {
  const int i = blockIdx.x * blockDim.x + threadIdx.x;
  if (i < N_NODES * HD) accum[i] = 0.f;
  if (i < N_NODES * HEADS) {
    segmax[i] = -3.0e38f;
    denom[i]  = 0.f;
  }
}

// Implicit augmented edge list: [src0,dst0] ++ [dst0,src0] ++ self loops.
__device__ __forceinline__ void edge_sd(const long long* __restrict__ ei,
                                        int e, int& s, int& d) {
  if (e < N_EDGES) {
    s = (int)ei[e];            d = (int)ei[N_EDGES + e];
  } else if (e < 2 * N_EDGES) {
    const int e0 = e - N_EDGES;
    s = (int)ei[N_EDGES + e0]; d = (int)ei[e0];
  } else {
    s = d = e - 2 * N_EDGES;
  }
}

__device__ __forceinline__ float leaky02(float v) {
  return v > 0.f ? v : 0.2f * v;
}

// float atomic max via signed-max / unsigned-min trick (init = -3e38)
__device__ __forceinline__ void atomicMaxFloat(float* addr, float v) {
  if (v >= 0.f) atomicMax((int*)addr, __float_as_int(v));
  else          atomicMin((unsigned int*)addr, __float_as_uint(v));
}

// ---------------------------------------------------------------------------
// Kernel 4: segment max of leaky-relu scores over dst  (one thread per edge)
// ---------------------------------------------------------------------------
__global__ void segmax_kernel(const long long* __restrict__ ei,
                              const float* __restrict__ alpha_src,
                              const float* __restrict__ alpha_dst,
                              float* __restrict__ segmax) {
  const int e = blockIdx.x * blockDim.x + threadIdx.x;
  if (e >= E_AUG) return;
  int s, d;
  edge_sd(ei, e, s, d);
  #pragma unroll
  for (int h = 0; h < HEADS; ++h) {
    const float sc = leaky02(alpha_src[s * HEADS + h] + alpha_dst[d * HEADS + h]);
    atomicMaxFloat(&segmax[d * HEADS + h], sc);
  }
}

// ---------------------------------------------------------------------------
// Kernel 5: segment sum of exp(score - max) over dst
// ---------------------------------------------------------------------------
__global__ void denom_kernel(const long long* __restrict__ ei,
                             const float* __restrict__ alpha_src,
                             const float* __restrict__ alpha_dst,
                             const float* __restrict__ segmax,
                             float* __restrict__ denom) {
  const int e = blockIdx.x * blockDim.x + threadIdx.x;
  if (e >= E_AUG) return;
  int s, d;
  edge_sd(ei, e, s, d);
  #pragma unroll
  for (int h = 0; h < HEADS; ++h) {
    const float sc = leaky02(alpha_src[s * HEADS + h] + alpha_dst[d * HEADS + h]);
    const float ex = expf(sc - segmax[d * HEADS + h]);
    atomicAdd(&denom[d * HEADS + h], ex);
  }
}

// ---------------------------------------------------------------------------
// Kernel 6: message scatter. One wave32 per edge; each lane handles 4 floats
// (float4) of the 128-float proj[src] row -> coalesced 512B gather per edge,
// weighted by attn, atomicAdd into accum[dst].
// ---------------------------------------------------------------------------
__global__ void scatter_kernel(const long long* __restrict__ ei,
                               const float* __restrict__ alpha_src,
                               const float* __restrict__ alpha_dst,
                               const float* __restrict__ segmax,
                               const float* __restrict__ denom,
                               const float* __restrict__ proj,
                               float* __restrict__ accum) {
  const int gid  = blockIdx.x * blockDim.x + threadIdx.x;
  const int e    = gid >> 5;
  const int lane = gid & 31;
  if (e >= E_AUG) return;
  int s, d;
  edge_sd(ei, e, s, d);

  const int h  = lane >> 3;   // lane*4 / 32 : which head this lane's 4 dims are in
  const float sc = leaky02(alpha_src[s * HEADS + h] + alpha_dst[d * HEADS + h]);
  const float ex = expf(sc - segmax[d * HEADS + h]);
  const float w  = ex / fmaxf(denom[d * HEADS + h], 1e-12f);

  const float4 p = ((const float4*)(proj + (size_t)s * HD))[lane];
  float* dst = accum + (size_t)d * HD + lane * 4;
  atomicAdd(dst + 0, p.x * w);
  atomicAdd(dst + 1, p.y * w);
  atomicAdd(dst + 2, p.z * w);
  atomicAdd(dst + 3, p.w * w);
}

// ---------------------------------------------------------------------------
// Kernel 7: out = elu(accum + b1) @ Wf + bf.  One wave32 per node, shfl reduce.
// ---------------------------------------------------------------------------
__global__ void final_kernel(const float* __restrict__ accum,
                             const float* __restrict__ b1,
                             const float* __restrict__ Wf,
                             const float* __restrict__ bf,
                             float* __restrict__ out) {
  const int gid  = blockIdx.x * blockDim.x + threadIdx.x;
  const int n    = gid >> 5;
  const int lane = gid & 31;
  if (n >= N_NODES) return;

  float sum = 0.f;
  #pragma unroll
  for (int j = 0; j < 4; ++j) {
    const int c = lane * 4 + j;
    float v = accum[(size_t)n * HD + c] + b1[c];
    v = v > 0.f ? v : (expf(v) - 1.f);   // ELU, alpha=1
    sum += v * Wf[c];
  }
  #pragma unroll
  for (int off = 16; off > 0; off >>= 1)
    sum += __shfl_xor(sum, off, 32);
  if (lane == 0) out[n] = sum + bf[0];
}

// ---------------------------------------------------------------------------
extern "C" void kernel_launch(void* const* d_in, const int* in_sizes, int n_in,
                              void* d_out, int out_size, void* d_ws, size_t ws_size,
                              hipStream_t stream) {
  (void)in_sizes; (void)n_in; (void)out_size; (void)ws_size;

  const float*     x     = (const float*)d_in[0];
  const long long* ei    = (const long long*)d_in[1];   // int64 [2, E]
  const float*     W1    = (const float*)d_in[2];
  const float*     a_src = (const float*)d_in[3];
  const float*     a_dst = (const float*)d_in[4];
  const float*     b1    = (const float*)d_in[5];
  const float*     Wf    = (const float*)d_in[6];
  const float*     bf    = (const float*)d_in[7];
  float*           out   = (float*)d_out;

  char* ws = (char*)d_ws;
  float* proj      = (float*)ws; ws += (size_t)N_NODES * HD * sizeof(float);     // 25.6 MB
  float* accum     = (float*)ws; ws += (size_t)N_NODES * HD * sizeof(float);     // 25.6 MB
  float* alpha_src = (float*)ws; ws += (size_t)N_NODES * HEADS * sizeof(float);
  float* alpha_dst = (float*)ws; ws += (size_t)N_NODES * HEADS * sizeof(float);
  float* segmax    = (float*)ws; ws += (size_t)N_NODES * HEADS * sizeof(float);
  float* denom     = (float*)ws; ws += (size_t)N_NODES * HEADS * sizeof(float);

  // 1) projection GEMM on matrix cores: 3125 M-tiles * 8 N-tiles, 1 wave each
  gemm_proj_kernel<<<(N_NODES / 16) * (HD / 16), 32, 0, stream>>>(x, W1, proj);

  // 2) attention logits
  {
    const int total = N_NODES * HEADS;
    alpha_kernel<<<(total + 255) / 256, 256, 0, stream>>>(proj, a_src, a_dst,
                                                          alpha_src, alpha_dst);
  }

  // 3) init accumulators
  {
    const int total = N_NODES * HD;
    init_kernel<<<(total + 255) / 256, 256, 0, stream>>>(accum, segmax, denom);
  }

  // 4) segment max
  segmax_kernel<<<(E_AUG + 255) / 256, 256, 0, stream>>>(ei, alpha_src, alpha_dst,
                                                         segmax);
  // 5) softmax denominator
  denom_kernel<<<(E_AUG + 255) / 256, 256, 0, stream>>>(ei, alpha_src, alpha_dst,
                                                        segmax, denom);
  // 6) weighted message scatter: one wave32 per edge
  {
    const long long threads = (long long)E_AUG * 32;
    scatter_kernel<<<(int)((threads + 255) / 256), 256, 0, stream>>>(
        ei, alpha_src, alpha_dst, segmax, denom, proj, accum);
  }
  // 7) ELU + final linear: one wave32 per node
  {
    const long long threads = (long long)N_NODES * 32;
    final_kernel<<<(int)((threads + 255) / 256), 256, 0, stream>>>(accum, b1, Wf,
                                                                   bf, out);
  }
}